// Dense_GCN_40355512713502
// MI455X (gfx1250) — compile-verified
//
#include <hip/hip_runtime.h>
#include <hip/hip_bf16.h>

typedef float v2f __attribute__((ext_vector_type(2)));
typedef float v8f __attribute__((ext_vector_type(8)));

__device__ __forceinline__ v8f wmma_f32_4(v2f a, v2f b, v8f c) {
    // D = A(16x4 f32) x B(4x16 f32) + C(16x16 f32)
    return __builtin_amdgcn_wmma_f32_16x16x4_f32(false, a, false, b, (short)0, c,
                                                 false, false);
}

// ---------------- degree / normalization ----------------

__global__ void deg_init(float* deg, int n) {
    int i = blockIdx.x * blockDim.x + threadIdx.x;
    if (i < n) deg[i] = 1.0f;  // self loop contributes 1 to every node's degree
}

__global__ void deg_count(const int* __restrict__ col, float* deg, int E) {
    int e = blockIdx.x * blockDim.x + threadIdx.x;
    if (e < E) unsafeAtomicAdd(&deg[col[e]], 1.0f);
}

__global__ void deg_rsqrt(float* deg, int n) {
    int i = blockIdx.x * blockDim.x + threadIdx.x;
    if (i < n) deg[i] = rsqrtf(fmaxf(deg[i], 1.0f));  // deg -> dinv in place
}

// ---------------- WMMA f32 GEMM: H[N,FOUT] = X[N,K] @ W[K,FOUT] ----------------
// One wave computes one 16x16 tile. A-frag: lane holds row M=lane%16,
// VGPR v + 16-lane half select K. B-frag: lane holds col N=lane%16.
// C/D: VGPR v, half h -> M = v + 8*h, N = lane%16.

template <int K, int FOUT>
__global__ void gemm_wmma(const float* __restrict__ X, const float* __restrict__ W,
                          float* __restrict__ H, int n) {
    constexpr int CT = FOUT / 16;  // column tiles (waves across FOUT)
    constexpr int RT = 8 / CT;     // row tiles per 256-thread block
    const int lane = threadIdx.x & 31;
    const int wave = threadIdx.x >> 5;
    const int ct = wave % CT;
    const int rt = wave / CT;
    const int m0 = (blockIdx.x * RT + rt) * 16;
    if (m0 >= n) return;  // wave-uniform: EXEC stays all-ones for WMMA
    const int n0 = ct * 16;
    const int half = lane >> 4;
    const int nc = lane & 15;
    const float* xrow = X + (size_t)(m0 + nc) * K;  // lane%16 == row within tile

    v8f acc = {};
#pragma unroll
    for (int k0 = 0; k0 < K; k0 += 4) {
        const int k = k0 + 2 * half;
        v2f a = *(const v2f*)&xrow[k];
        v2f b;
        b.x = W[(size_t)(k + 0) * FOUT + n0 + nc];
        b.y = W[(size_t)(k + 1) * FOUT + n0 + nc];
        acc = wmma_f32_4(a, b, acc);
    }
#pragma unroll
    for (int v = 0; v < 8; ++v) {
        const int m = m0 + v + 8 * half;
        H[(size_t)m * FOUT + n0 + nc] = acc[v];
    }
}

// ---------------- message passing ----------------

// agg[i][f] = dinv[i]^2 * h[i][f]   (self-loop term; also initializes agg)
template <int F>
__global__ void self_init(const float* __restrict__ h, const float* __restrict__ dinv,
                          float* __restrict__ agg, int n) {
    unsigned tid = blockIdx.x * blockDim.x + threadIdx.x;
    unsigned total = (unsigned)n * F;
    if (tid >= total) return;
    int i = tid / F;
    float d = dinv[i];
    agg[tid] = d * d * h[tid];
}

// agg[col][f] += dinv[row]*dinv[col] * h[row][f]  for every edge
template <int F>
__global__ void edge_scatter(const int* __restrict__ row, const int* __restrict__ col,
                             const float* __restrict__ dinv, const float* __restrict__ h,
                             float* __restrict__ agg, int E) {
    unsigned tid = blockIdx.x * blockDim.x + threadIdx.x;
    int e = (int)(tid / F);
    int f = (int)(tid & (F - 1));
    if (e >= E) return;
    int r = row[e];
    int c = col[e];
    float nrm = dinv[r] * dinv[c];
    unsafeAtomicAdd(&agg[(size_t)c * F + f], nrm * h[(size_t)r * F + f]);
}

// f[i][f] = relu(agg[i][f] + b[f])
template <int F>
__global__ void bias_relu(const float* __restrict__ agg, const float* __restrict__ b,
                          float* __restrict__ f, int n) {
    unsigned tid = blockIdx.x * blockDim.x + threadIdx.x;
    unsigned total = (unsigned)n * F;
    if (tid >= total) return;
    f[tid] = fmaxf(agg[tid] + b[tid & (F - 1)], 0.0f);
}

// ---------------- dense head: out = relu([f1|f2|f3] @ Wfc + bfc) ----------------
// K = 112 (64 + 32 + 16), FOUT = 16. Segment boundaries (64, 96) are multiples
// of the K-step (4), so the buffer selection is wave-uniform per k0.

__global__ void fc_wmma(const float* __restrict__ f1, const float* __restrict__ f2,
                        const float* __restrict__ f3, const float* __restrict__ Wfc,
                        const float* __restrict__ bfc, float* __restrict__ out, int n) {
    const int lane = threadIdx.x & 31;
    const int wave = threadIdx.x >> 5;
    const int m0 = (blockIdx.x * 8 + wave) * 16;
    if (m0 >= n) return;
    const int half = lane >> 4;
    const int nc = lane & 15;
    const int mr = m0 + nc;  // row this lane supplies for the A fragment

    v8f acc = {};
#pragma unroll
    for (int k0 = 0; k0 < 112; k0 += 4) {
        const int k = k0 + 2 * half;
        v2f a;
        if (k0 < 64)
            a = *(const v2f*)&f1[(size_t)mr * 64 + k];
        else if (k0 < 96)
            a = *(const v2f*)&f2[(size_t)mr * 32 + (k - 64)];
        else
            a = *(const v2f*)&f3[(size_t)mr * 16 + (k - 96)];
        v2f b;
        b.x = Wfc[(k + 0) * 16 + nc];
        b.y = Wfc[(k + 1) * 16 + nc];
        acc = wmma_f32_4(a, b, acc);
    }
    const float bias = bfc[nc];
#pragma unroll
    for (int v = 0; v < 8; ++v) {
        const int m = m0 + v + 8 * half;
        out[(size_t)m * 16 + nc] = fmaxf(acc[v] + bias, 0.0f);
    }
}

// ---------------- launch ----------------

extern "C" void kernel_launch(void* const* d_in, const int* in_sizes, int n_in,
                              void* d_out, int out_size, void* d_ws, size_t ws_size,
                              hipStream_t stream) {
    const int* edges = (const int*)d_in[0];
    const float* X   = (const float*)d_in[1];
    const float* W1  = (const float*)d_in[2];
    const float* b1  = (const float*)d_in[3];
    const float* W2  = (const float*)d_in[4];
    const float* b2  = (const float*)d_in[5];
    const float* W3  = (const float*)d_in[6];
    const float* b3  = (const float*)d_in[7];
    const float* Wfc = (const float*)d_in[8];
    const float* bfc = (const float*)d_in[9];
    float* out = (float*)d_out;

    const int E = in_sizes[0] / 2;
    const int N = in_sizes[1] / 128;
    const int* row = edges;      // edges[0]
    const int* col = edges + E;  // edges[1]

    // workspace layout (floats)
    float* ws = (float*)d_ws;
    float* dinv = ws;                       // N
    float* h1   = dinv + N;                 // N*64 (f1 aliases h1 after agg)
    float* agg1 = h1 + (size_t)N * 64;      // N*64
    float* h2   = agg1 + (size_t)N * 64;    // N*32
    float* agg2 = h2 + (size_t)N * 32;      // N*32
    float* h3   = agg2 + (size_t)N * 32;    // N*16
    float* agg3 = h3 + (size_t)N * 16;      // N*16
    float* f1 = h1;
    float* f2 = h2;
    float* f3 = h3;

    const int T = 256;
    auto cdiv = [](long long a, long long b) { return (int)((a + b - 1) / b); };

    // normalization
    deg_init<<<cdiv(N, T), T, 0, stream>>>(dinv, N);
    deg_count<<<cdiv(E, T), T, 0, stream>>>(col, dinv, E);
    deg_rsqrt<<<cdiv(N, T), T, 0, stream>>>(dinv, N);

    // layer 1: 128 -> 64
    gemm_wmma<128, 64><<<cdiv(N, 16 * 2), T, 0, stream>>>(X, W1, h1, N);
    self_init<64><<<cdiv((long long)N * 64, T), T, 0, stream>>>(h1, dinv, agg1, N);
    edge_scatter<64><<<cdiv((long long)E * 64, T), T, 0, stream>>>(row, col, dinv, h1, agg1, E);
    bias_relu<64><<<cdiv((long long)N * 64, T), T, 0, stream>>>(agg1, b1, f1, N);

    // layer 2: 64 -> 32
    gemm_wmma<64, 32><<<cdiv(N, 16 * 4), T, 0, stream>>>(f1, W2, h2, N);
    self_init<32><<<cdiv((long long)N * 32, T), T, 0, stream>>>(h2, dinv, agg2, N);
    edge_scatter<32><<<cdiv((long long)E * 32, T), T, 0, stream>>>(row, col, dinv, h2, agg2, E);
    bias_relu<32><<<cdiv((long long)N * 32, T), T, 0, stream>>>(agg2, b2, f2, N);

    // layer 3: 32 -> 16
    gemm_wmma<32, 16><<<cdiv(N, 16 * 8), T, 0, stream>>>(f2, W3, h3, N);
    self_init<16><<<cdiv((long long)N * 16, T), T, 0, stream>>>(h3, dinv, agg3, N);
    edge_scatter<16><<<cdiv((long long)E * 16, T), T, 0, stream>>>(row, col, dinv, h3, agg3, E);
    bias_relu<16><<<cdiv((long long)N * 16, T), T, 0, stream>>>(agg3, b3, f3, N);

    // dense head: [N,112] @ [112,16] + bias, relu
    fc_wmma<<<cdiv(N, 16 * 8), T, 0, stream>>>(f1, f2, f3, Wfc, bfc, out, N);
}